// RgbContProj_80109730005088
// MI455X (gfx1250) — compile-verified
//
#include <hip/hip_runtime.h>
#include <hip/hip_bf16.h>

typedef float fx2 __attribute__((ext_vector_type(2)));
typedef float fx8 __attribute__((ext_vector_type(8)));
typedef unsigned int u32x4 __attribute__((ext_vector_type(4)));
typedef int i32x4 __attribute__((ext_vector_type(4)));
typedef int i32x8 __attribute__((ext_vector_type(8)));

#define B_SZ     8
#define N_PTS    1024
#define GHW      4096      // 64*64
#define CELLS_PB 128       // cells (h,w) per block: 8 waves * 16 rows
#define WELL_R   1.0f
#define BETA_K   100.0f

#define HAVE_TDM (__has_builtin(__builtin_amdgcn_tensor_load_to_lds))

__device__ __forceinline__ float depth_of(float x, float y, float z1,
                                          float gx, float gy) {
    float kx = (fabsf(x - gx) <= WELL_R) ? 1.0f : 10.0f;
    float ky = (fabsf(y - gy) <= WELL_R) ? 1.0f : 10.0f;
    float gv = kx * ky * z1;
    return fminf(fmaxf(gv, 0.0f), 10.0f);
}

#if HAVE_TDM
// 1-D TDM copy: nelem f32 from global gsrc -> LDS byte offset lds_off.
// D# layout per CDNA5 ISA 8.3/8.4: group0 {count=1, lds_addr, global_addr,
// type=2}; group1 {data_size=4B, tensor_dim0=tile_dim0=nelem, dims1=1,
// stride0=nelem}; groups 2/3 zero (tile_dim3/4 unused).
__device__ __forceinline__ void tdm_copy_f32(unsigned lds_off,
                                             const void* gsrc,
                                             unsigned nelem) {
    unsigned long long ga = (unsigned long long)gsrc;
    u32x4 g0;
    g0[0] = 1u;                                        // count=1, user mode
    g0[1] = lds_off;                                   // lds_addr (bytes)
    g0[2] = (unsigned)(ga & 0xFFFFFFFFu);              // global_addr[31:0]
    g0[3] = (unsigned)((ga >> 32) & 0x01FFFFFFu)       // global_addr[56:32]
          | (2u << 30);                                // type=2 ("image")
    i32x8 g1;
    g1[0] = (int)(2u << 16);                           // data_size=2 (4B)
    g1[1] = (int)((nelem & 0xFFFFu) << 16);            // tensor_dim0[15:0]
    g1[2] = (int)((nelem >> 16) | (1u << 16));         // dim0[31:16] | dim1=1
    g1[3] = (int)((nelem & 0xFFFFu) << 16);            // tile_dim0
    g1[4] = 1;                                         // tile_dim1=1
    g1[5] = (int)nelem;                                // dim0_stride[31:0]
    g1[6] = 0;
    g1[7] = 0;
    i32x4 gz = {};
#if __clang_major__ >= 23
    i32x8 z8 = {};
    __builtin_amdgcn_tensor_load_to_lds(g0, g1, gz, gz, z8, 0);
#else
    __builtin_amdgcn_tensor_load_to_lds(g0, g1, gz, gz, 0);
#endif
}
#endif

__global__ __launch_bounds__(256)
void RgbContProj_kernel(const float* __restrict__ pcl,
                        const float* __restrict__ feat,
                        float* __restrict__ proj,   // (B,64,64,3)
                        float* __restrict__ prob,   // (B,N,64,64)
                        float* __restrict__ maskout)// (B,64,64)
{
    __shared__ float s_pcl[N_PTS * 3];
    __shared__ float s_feat[N_PTS * 3];
    __shared__ float s_bcol[N_PTS * 4 + 4];  // {f0,f1,f2,1} per n + zero slot
    __shared__ float s_m[CELLS_PB];
    __shared__ float s_S[CELLS_PB];
    __shared__ float s_cnt[CELLS_PB];
    __shared__ float s_invden[CELLS_PB];
    __shared__ float s_dtile[8][256];        // one 16x16 f32 D tile per wave

    const int tid  = threadIdx.x;
    const int wv   = tid >> 5;
    const int lane = tid & 31;
    const int row  = lane & 15;   // M row (A) == N col (B)
    const int half = lane >> 4;   // K half: lanes 0-15 -> K0,K1 ; 16-31 -> K2,K3

    const int bid      = blockIdx.x;
    const int b        = bid >> 5;          // 32 blocks per batch
    const int cellbase = (bid & 31) << 7;   // 128 cells per block

    // ---- stage pcl[b], feat[b] into LDS ----
#if HAVE_TDM
    if (wv == 0) {
        const unsigned pcl_off  = (unsigned)(unsigned long long)(void*)s_pcl;
        const unsigned feat_off = (unsigned)(unsigned long long)(void*)s_feat;
        tdm_copy_f32(pcl_off,  pcl  + (size_t)b * N_PTS * 3, N_PTS * 3);
        tdm_copy_f32(feat_off, feat + (size_t)b * N_PTS * 3, N_PTS * 3);
        __builtin_amdgcn_s_wait_tensorcnt(0);
    }
#else
    {
        const float4* ps = (const float4*)(pcl  + (size_t)b * N_PTS * 3);
        const float4* fs = (const float4*)(feat + (size_t)b * N_PTS * 3);
        float4* pd = (float4*)s_pcl;
        float4* fd = (float4*)s_feat;
        for (int i = tid; i < 768; i += 256) { pd[i] = ps[i]; fd[i] = fs[i]; }
    }
#endif
    __syncthreads();

    // branchless B-column table: s_bcol[4n..4n+3] = {f0,f1,f2,1}; [4096] = 0
    for (int n = tid; n < N_PTS; n += 256) {
        s_bcol[n * 4 + 0] = s_feat[n * 3 + 0];
        s_bcol[n * 4 + 1] = s_feat[n * 3 + 1];
        s_bcol[n * 4 + 2] = s_feat[n * 3 + 2];
        s_bcol[n * 4 + 3] = 1.0f;
    }
    if (tid == 0) s_bcol[N_PTS * 4] = 0.0f;
    __syncthreads();

    const int   cell = wv * 16 + row;
    const int   hw   = cellbase + cell;
    const float gx   = (float)(hw >> 6);   // grid x == h  (meshgrid 'ij')
    const float gy   = (float)(hw & 63);   // grid y == w

    // ---- pass 1: max logit over n (lane covers n = nb + 2*half + {0,1}) ----
    float m = -3.4e38f;
    for (int nb = 0; nb < N_PTS; nb += 4) {
#pragma unroll
        for (int j = 0; j < 2; ++j) {
            const int n = nb + 2 * half + j;
            const float x  = s_pcl[n * 3 + 0];
            const float y  = s_pcl[n * 3 + 1];
            const float z1 = s_pcl[n * 3 + 2] + 1.0f;
            const float d  = depth_of(x, y, z1, gx, gy);
            m = fmaxf(m, BETA_K / (d + 1e-5f));
        }
    }
    m = fmaxf(m, __shfl_xor(m, 16));   // combine the two K half-slices

    // ---- pass 2: WMMA-fused accumulation ----
    // A(16x4)[row][k] = e*mask ; B(4x16)[k][col] = {feat0,feat1,feat2,1,0..}
    // D(16x16): cols 0..2 = sum e*mask*feat_c ; col 3 = Smask
    const unsigned bbase = (row < 4) ? (unsigned)row : (unsigned)(N_PTS * 4);
    const unsigned bstep = (row < 4) ? 4u : 0u;
    fx8 acc = {};
    float Sloc = 0.0f, cntloc = 0.0f;
    for (int nb = 0; nb < N_PTS; nb += 4) {
        fx2 a, bm;
#pragma unroll
        for (int j = 0; j < 2; ++j) {
            const int n = nb + 2 * half + j;
            const float x  = s_pcl[n * 3 + 0];
            const float y  = s_pcl[n * 3 + 1];
            const float z1 = s_pcl[n * 3 + 2] + 1.0f;
            const float d  = depth_of(x, y, z1, gx, gy);
            const float l  = BETA_K / (d + 1e-5f);
            const float e  = __expf(l - m);
            const float mk = (d != 10.0f) ? 1.0f : 0.0f;
            a[j] = e * mk;
            Sloc   += e;
            cntloc += mk;
            bm[j] = s_bcol[bbase + (unsigned)n * bstep];  // branchless
        }
        acc = __builtin_amdgcn_wmma_f32_16x16x4_f32(
                  /*neg_a=*/false, a, /*neg_b=*/false, bm,
                  /*c_mod=*/(short)0, acc,
                  /*reuse_a=*/false, /*reuse_b=*/false);
    }
    const float S_row   = Sloc   + __shfl_xor(Sloc, 16);
    const float cnt_row = cntloc + __shfl_xor(cntloc, 16);
    if (half == 0) {
        s_m[cell]   = m;
        s_S[cell]   = S_row;
        s_cnt[cell] = cnt_row;
    }
    // spill D tile: lane holds D[r + 8*half][row] in acc[r]
#pragma unroll
    for (int r = 0; r < 8; ++r)
        s_dtile[wv][(r + 8 * half) * 16 + row] = acc[r];
    __syncthreads();

    // ---- finalize per cell: invden, mask, proj_feat ----
    if (tid < CELLS_PB) {
        const int c2  = tid;
        const int wvc = c2 >> 4;
        const int rw  = c2 & 15;
        const float Smask = s_dtile[wvc][rw * 16 + 3];
        const float inv   = 1.0f / (Smask + 1e-8f * s_S[c2]);
        const float mkf   = (s_cnt[c2] != 0.0f) ? 1.0f : 0.0f;
        s_invden[c2] = inv;
        const int hw2 = cellbase + c2;
        maskout[b * GHW + hw2] = mkf;
        const float sc = inv * mkf;
        float* pf = proj + ((size_t)b * GHW + hw2) * 3;
        pf[0] = s_dtile[wvc][rw * 16 + 0] * sc;
        pf[1] = s_dtile[wvc][rw * 16 + 1] * sc;
        pf[2] = s_dtile[wvc][rw * 16 + 2] * sc;
    }
    __syncthreads();

    // ---- pass 3: stream prob, coalesced (128 consecutive lanes per n) ----
    const int   c3   = tid & 127;
    const int   nh   = tid >> 7;       // two n-streams of 512 each
    const int   hw3  = cellbase + c3;
    const float gx3  = (float)(hw3 >> 6);
    const float gy3  = (float)(hw3 & 63);
    const float m3   = s_m[c3];
    const float inv3 = s_invden[c3];
    float* pp = prob + (size_t)b * N_PTS * GHW + hw3;
    const int n0 = nh * 512;
    for (int n = n0; n < n0 + 512; ++n) {
        const float x  = s_pcl[n * 3 + 0];
        const float y  = s_pcl[n * 3 + 1];
        const float z1 = s_pcl[n * 3 + 2] + 1.0f;
        const float d  = depth_of(x, y, z1, gx3, gy3);
        const float l  = BETA_K / (d + 1e-5f);
        const float e  = __expf(l - m3);
        const float mk = (d != 10.0f) ? 1.0f : 0.0f;
        pp[(size_t)n * GHW] = e * mk * inv3;
    }
}

extern "C" void kernel_launch(void* const* d_in, const int* in_sizes, int n_in,
                              void* d_out, int out_size, void* d_ws, size_t ws_size,
                              hipStream_t stream) {
    const float* pcl  = (const float*)d_in[0];   // (8,1024,3) f32
    const float* feat = (const float*)d_in[1];   // (8,1024,3) f32
    float* out  = (float*)d_out;
    float* proj = out;                                   // 8*64*64*3
    float* prob = proj + (size_t)B_SZ * GHW * 3;         // 8*1024*64*64
    float* mask = prob + (size_t)B_SZ * N_PTS * GHW;     // 8*64*64

    RgbContProj_kernel<<<B_SZ * (GHW / CELLS_PB), 256, 0, stream>>>(
        pcl, feat, proj, prob, mask);
}